// RealAgnosticInteractionBlock_21766894256490
// MI455X (gfx1250) — compile-verified
//
#include <hip/hip_runtime.h>
#include <math.h>
#include <stdint.h>

typedef float v2f __attribute__((ext_vector_type(2)));
typedef float v8f __attribute__((ext_vector_type(8)));

#define N_NODES 5000
#define N_EDGES 80000
#define CCH 128
#define KATTR 10
#define NBAS 8
#define SHD 16

__device__ __forceinline__ v8f wmma_f32(v2f a, v2f b, v8f c) {
  // D = A(16x4 f32) * B(4x16 f32) + C(16x16 f32)
  return __builtin_amdgcn_wmma_f32_16x16x4_f32(false, a, false, b, (short)0, c, false, false);
}

__device__ __forceinline__ float silu_f(float x) { return x / (1.0f + expf(-x)); }

// ---------------------------------------------------------------------------
// Kernel 1: x_up = node_feats @ W_up / sqrt(128)
// 8 waves/block, each wave owns one 16-wide column tile; 313 row tiles.
// ---------------------------------------------------------------------------
__global__ void __launch_bounds__(256) k_linear_up(const float* __restrict__ nf,
                                                   const float* __restrict__ Wup,
                                                   float* __restrict__ xup) {
  const int lane = threadIdx.x & 31;
  const int wave = threadIdx.x >> 5;
  const int mbase = blockIdx.x * 16;
  const int vt = wave * 16;
  const int mrow = lane & 15;
  const int ncol = lane & 15;
  const int koff = (lane >> 4) << 1;
  const int half8 = (lane >> 4) << 3;

  int grow = mbase + mrow; if (grow > N_NODES - 1) grow = N_NODES - 1;
  const float* Arow  = nf + (size_t)grow * CCH + koff;
  const float* Bbase = Wup + koff * CCH + vt + ncol;

  v8f acc = {};
  for (int k0 = 0; k0 < CCH; k0 += 4) {
    v2f a; a.x = Arow[k0];        a.y = Arow[k0 + 1];
    v2f b; b.x = Bbase[k0 * CCH]; b.y = Bbase[(k0 + 1) * CCH];
    acc = wmma_f32(a, b, acc);
  }
  const float s = 0.08838834764831845f; // 1/sqrt(128)
#pragma unroll
  for (int j = 0; j < 8; ++j) {
    const int r = mbase + j + half8;
    if (r < N_NODES) xup[(size_t)r * CCH + vt + ncol] = acc[j] * s;
  }
}

// ---------------------------------------------------------------------------
// Kernel 2: fused radial MLP (WMMA) + edge message + atomic scatter.
// One block = 16 edges, 4 waves. tp weights [16,512] stay in LDS.
// rW1/rW2 are staged into LDS via CDNA5 async LDS copies, overlapped with
// layer 0; B-fragments for layers 1-2 then come from LDS (padded stride 80).
// ---------------------------------------------------------------------------
__global__ void __launch_bounds__(128) k_radial_message(
    const float* __restrict__ ef, const float* __restrict__ ea, const int* __restrict__ ei,
    const float* __restrict__ rW0, const float* __restrict__ rW1,
    const float* __restrict__ rW2, const float* __restrict__ rW3,
    const float* __restrict__ xup, float* __restrict__ mbuf) {
  __shared__ float hA[16][64];
  __shared__ float hB[16][64];
  __shared__ float tp[16][512];
  __shared__ float sW[2][64][80];   // staged rW1, rW2 (row stride 80 floats)

  const int tid = threadIdx.x;
  const int lane = tid & 31;
  const int wave = tid >> 5;
  const int e0 = blockIdx.x * 16;
  const int mrow = lane & 15;
  const int ncol = lane & 15;
  const int koff = (lane >> 4) << 1;
  const int half8 = (lane >> 4) << 3;

  // ---- issue async global->LDS staging of rW1/rW2 (16 x b128 per lane) ----
  {
    const unsigned lbase = (unsigned)(uintptr_t)&sW[0][0][0];
#pragma unroll
    for (int it = 0; it < 8; ++it) {
      const int idx = tid + it * 128;                 // b128 chunk id (16 per 64-f row)
      const unsigned laddr = lbase + (unsigned)((idx >> 4) * 320 + (idx & 15) * 16);
      const float* g = rW1 + idx * 4;
      asm volatile("global_load_async_to_lds_b128 %0, %1, off"
                   :: "v"(laddr), "v"(g) : "memory");
    }
    const unsigned lbase2 = lbase + 64u * 320u;
#pragma unroll
    for (int it = 0; it < 8; ++it) {
      const int idx = tid + it * 128;
      const unsigned laddr = lbase2 + (unsigned)((idx >> 4) * 320 + (idx & 15) * 16);
      const float* g = rW2 + idx * 4;
      asm volatile("global_load_async_to_lds_b128 %0, %1, off"
                   :: "v"(laddr), "v"(g) : "memory");
    }
  }

  // layer 0: [16,8] @ rW0[8,64], silu, /sqrt(8)  (overlaps the async copies)
  {
    int ge = e0 + mrow; if (ge > N_EDGES - 1) ge = N_EDGES - 1;
    const float* Arow  = ef + (size_t)ge * NBAS + koff;
    const int vt = wave * 16;
    const float* Bbase = rW0 + koff * 64 + vt + ncol;
    v8f acc = {};
    for (int k0 = 0; k0 < 8; k0 += 4) {
      v2f a; a.x = Arow[k0];       a.y = Arow[k0 + 1];
      v2f b; b.x = Bbase[k0 * 64]; b.y = Bbase[(k0 + 1) * 64];
      acc = wmma_f32(a, b, acc);
    }
#pragma unroll
    for (int j = 0; j < 8; ++j)
      hA[j + half8][vt + ncol] = silu_f(acc[j] * 0.3535533905932738f);
  }
  asm volatile("s_wait_asynccnt 0" ::: "memory");
  __syncthreads();

  // layer 1: hA @ rW1[64,64] -> hB, silu, /sqrt(64)  (B from LDS)
  {
    const int vt = wave * 16;
    v8f acc = {};
    for (int k0 = 0; k0 < 64; k0 += 4) {
      v2f a; a.x = hA[mrow][k0 + koff];          a.y = hA[mrow][k0 + koff + 1];
      v2f b; b.x = sW[0][k0 + koff][vt + ncol];  b.y = sW[0][k0 + koff + 1][vt + ncol];
      acc = wmma_f32(a, b, acc);
    }
#pragma unroll
    for (int j = 0; j < 8; ++j)
      hB[j + half8][vt + ncol] = silu_f(acc[j] * 0.125f);
  }
  __syncthreads();
  // layer 2: hB @ rW2[64,64] -> hA, silu, /sqrt(64)  (B from LDS)
  {
    const int vt = wave * 16;
    v8f acc = {};
    for (int k0 = 0; k0 < 64; k0 += 4) {
      v2f a; a.x = hB[mrow][k0 + koff];          a.y = hB[mrow][k0 + koff + 1];
      v2f b; b.x = sW[1][k0 + koff][vt + ncol];  b.y = sW[1][k0 + koff + 1][vt + ncol];
      acc = wmma_f32(a, b, acc);
    }
#pragma unroll
    for (int j = 0; j < 8; ++j)
      hA[j + half8][vt + ncol] = silu_f(acc[j] * 0.125f);
  }
  __syncthreads();
  // layer 3: hA @ rW3[64,512] -> tp, /sqrt(64), no activation. 8 tiles/wave.
  for (int t = 0; t < 8; ++t) {
    const int vt = wave * 128 + t * 16;
    const float* Bbase = rW3 + koff * 512 + vt + ncol;
    v8f acc = {};
    for (int k0 = 0; k0 < 64; k0 += 4) {
      v2f a; a.x = hA[mrow][k0 + koff]; a.y = hA[mrow][k0 + koff + 1];
      v2f b; b.x = Bbase[k0 * 512];     b.y = Bbase[(k0 + 1) * 512];
      acc = wmma_f32(a, b, acc);
    }
#pragma unroll
    for (int j = 0; j < 8; ++j)
      tp[j + half8][vt + ncol] = acc[j] * 0.125f;
  }
  __syncthreads();

  // message: mji[c,m] = x_up[snd,c] * tp[e][l(m)*128+c] * ea[e,m]; scatter to rcv
  for (int el = 0; el < 16; ++el) {
    const int e = e0 + el;
    const int snd = ei[e];
    const int rcv = ei[N_EDGES + e];
    const float* xs = xup + (size_t)snd * CCH;
    float* mb = mbuf + (size_t)rcv * (CCH * SHD);
    for (int idx = tid; idx < CCH * SHD; idx += 128) {
      const int c = idx >> 4;
      const int m = idx & 15;
      const int l = (m >= 9) ? 3 : (m >= 4) ? 2 : (m >= 1) ? 1 : 0;
      const float v = xs[c] * tp[el][l * CCH + c] * ea[(size_t)e * SHD + m];
      unsafeAtomicAdd(mb + idx, v);
    }
  }
}

// ---------------------------------------------------------------------------
// Kernel 3: per-l linear (+/16, /sqrt(128)) then skip TP with node_attrs.
// Rows = (node, m-component) pairs for one l; 16-row tiles, 4 waves/block.
// ---------------------------------------------------------------------------
__global__ void __launch_bounds__(128) k_out(
    const float* __restrict__ mbuf, const float* __restrict__ nattr,
    const float* __restrict__ Wlin, const float* __restrict__ Wskip,
    float* __restrict__ out) {
  __shared__ float mlT[16][CCH];
  __shared__ float attrT[16][KATTR];

  const int tid = threadIdx.x;
  const int lane = tid & 31;
  const int wave = tid >> 5;
  const int mrow = lane & 15;
  const int ncol = lane & 15;
  const int koff = (lane >> 4) << 1;
  const int half8 = (lane >> 4) << 3;

  // block -> (l, row tile). Tiles: 313 / 938 / 1563 / 2188.
  int bid = blockIdx.x, l, tile;
  if (bid < 313)       { l = 0; tile = bid; }
  else if (bid < 1251) { l = 1; tile = bid - 313; }
  else if (bid < 2814) { l = 2; tile = bid - 1251; }
  else                 { l = 3; tile = bid - 2814; }
  const int Ld = 2 * l + 1;
  const int offs = (l == 0) ? 0 : (l == 1) ? 1 : (l == 2) ? 4 : 9;
  const int rows = N_NODES * Ld;
  const int rowbase = tile * 16;

  if (tid < 16 * KATTR) {
    const int r = tid / KATTR, k = tid % KATTR;
    int grow = rowbase + r; if (grow > rows - 1) grow = rows - 1;
    const int n = grow / Ld;
    attrT[r][k] = nattr[n * KATTR + k];
  }

  // per-lane A-row mapping: A[row,u] = mbuf[n, u, offs+i]
  int grow = rowbase + mrow; if (grow > rows - 1) grow = rows - 1;
  const int n = grow / Ld;
  const int mm = offs + (grow - n * Ld);
  const float* Abase = mbuf + (size_t)n * (CCH * SHD) + mm;

  // phase 1: ml = (A @ W_lin[l]) / sqrt(128) / 16  -> LDS
  const float* WlinL = Wlin + l * CCH * CCH;
  for (int t = 0; t < 2; ++t) {
    const int vt = (wave * 2 + t) * 16;
    const float* Bbase = WlinL + koff * CCH + vt + ncol;
    v8f acc = {};
    for (int u0 = 0; u0 < CCH; u0 += 4) {
      v2f a; a.x = Abase[(u0 + koff) * SHD]; a.y = Abase[(u0 + koff + 1) * SHD];
      v2f b; b.x = Bbase[u0 * CCH];          b.y = Bbase[(u0 + 1) * CCH];
      acc = wmma_f32(a, b, acc);
    }
#pragma unroll
    for (int j = 0; j < 8; ++j)
      mlT[j + half8][vt + ncol] = acc[j] * 0.005524271728019903f; // 1/(16*sqrt(128))
  }
  __syncthreads();

  // phase 2: out = sum_k attr[n,k] * (ml @ Wskip[l,:,k,:]) / sqrt(1280)
  const float* WskL = Wskip + (size_t)l * CCH * KATTR * CCH;
  const float scale2 = 0.027950849718747374f; // 1/sqrt(1280)
  for (int t = 0; t < 2; ++t) {
    const int vt = (wave * 2 + t) * 16;
    v8f acc = {};
    for (int k = 0; k < KATTR; ++k) {
      const float* Bbase = WskL + koff * (KATTR * CCH) + k * CCH + vt + ncol;
      v8f tk = {};
      for (int u0 = 0; u0 < CCH; u0 += 4) {
        v2f a; a.x = mlT[mrow][u0 + koff];      a.y = mlT[mrow][u0 + koff + 1];
        v2f b; b.x = Bbase[u0 * (KATTR * CCH)]; b.y = Bbase[(u0 + 1) * (KATTR * CCH)];
        tk = wmma_f32(a, b, tk);
      }
#pragma unroll
      for (int j = 0; j < 8; ++j)
        acc[j] += attrT[j + half8][k] * tk[j];
    }
#pragma unroll
    for (int j = 0; j < 8; ++j) {
      const int r = rowbase + j + half8;
      if (r < rows) {
        const int nn = r / Ld;
        const int ii = r - nn * Ld;
        out[(size_t)nn * (CCH * SHD) + (size_t)(vt + ncol) * SHD + offs + ii] = acc[j] * scale2;
      }
    }
  }
}

// ---------------------------------------------------------------------------
extern "C" void kernel_launch(void* const* d_in, const int* in_sizes, int n_in,
                              void* d_out, int out_size, void* d_ws, size_t ws_size,
                              hipStream_t stream) {
  const float* node_attrs = (const float*)d_in[0];
  const float* node_feats = (const float*)d_in[1];
  const float* edge_attrs = (const float*)d_in[2];
  const float* edge_feats = (const float*)d_in[3];
  const int*   edge_index = (const int*)d_in[4];
  const float* W_up   = (const float*)d_in[5];
  const float* rW0    = (const float*)d_in[6];
  const float* rW1    = (const float*)d_in[7];
  const float* rW2    = (const float*)d_in[8];
  const float* rW3    = (const float*)d_in[9];
  const float* W_lin  = (const float*)d_in[10];
  const float* W_skip = (const float*)d_in[11];
  float* out = (float*)d_out;

  const size_t xup_bytes = ((size_t)N_NODES * CCH * sizeof(float) + 255) & ~(size_t)255;
  float* x_up  = (float*)d_ws;
  float* m_buf = (float*)((char*)d_ws + xup_bytes);

  hipMemsetAsync(m_buf, 0, (size_t)N_NODES * CCH * SHD * sizeof(float), stream);
  hipLaunchKernelGGL(k_linear_up, dim3(313), dim3(256), 0, stream,
                     node_feats, W_up, x_up);
  hipLaunchKernelGGL(k_radial_message, dim3(5000), dim3(128), 0, stream,
                     edge_feats, edge_attrs, edge_index, rW0, rW1, rW2, rW3, x_up, m_buf);
  hipLaunchKernelGGL(k_out, dim3(5002), dim3(128), 0, stream,
                     m_buf, node_attrs, W_lin, W_skip, out);
}